// NET_18193481466022
// MI455X (gfx1250) — compile-verified
//
#include <hip/hip_runtime.h>
#include <hip/hip_bf16.h>

typedef __attribute__((ext_vector_type(2))) float v2f;
typedef __attribute__((ext_vector_type(4))) float v4f;
typedef __attribute__((ext_vector_type(8))) float v8f;

#define HW      2304        // 48*48
#define QSTR    27648       // 4*3*2304  (stride between q-planes of core)
#define NB      8           // N
#define NBATCH  4           // B
#define PREDI_ELEMS (NBATCH * NB * 3 * HW)   // 221184

// -------- per-K term: sum_m a[m] * (sum_l b[l] * f[m+P][l+P]) --------
// K,P,CUR compile-time so the f[7][7] register window indices stay constant.
template<int K, int P, int CUR>
__device__ __forceinline__ float s_term(const float* __restrict__ core,
                                        int bct, const float (&f)[7][7]) {
  float bb[K];
#pragma unroll
  for (int l = 0; l < K; ++l)
    bb[l] = core[bct + (16 + CUR + l) * QSTR];
  float sum = 0.f;
#pragma unroll
  for (int m = 0; m < K; ++m) {
    float a = core[bct + (CUR + m) * QSTR];
    float inner = 0.f;
#pragma unroll
    for (int l = 0; l < K; ++l)
      inner = fmaf(bb[l], f[m + P][l + P], inner);
    sum = fmaf(a, inner, sum);
  }
  return sum;
}

// ---------------- Kernel 1: fused kernel-conv -> pred_img_i ----------------
// One thread per (pixel, t): 768 blocks x 288 threads (48 cols x 2 rows x 3 t).
// Every core element is read exactly once, coalesced along x.
__global__ __launch_bounds__(288)
void kconv_kernel(const float* __restrict__ frames,
                  const float* __restrict__ core,
                  const float* __restrict__ kw,
                  float* __restrict__ predI) {
  const int tid  = threadIdx.x;
  const int bn   = blockIdx.x / 24;         // b*8 + n
  const int tile = blockIdx.x % 24;
  const int y0   = tile * 2;

  // stage frames halo tile: 4 channels x 8 rows x 54 cols (6.9 KB)
  __shared__ float fs[4][8][54];
  for (int e = tid; e < 4 * 8 * 54; e += 288) {
    int c   = e / (8 * 54);
    int rem = e % (8 * 54);
    int row = rem / 54, col = rem % 54;
    int gy = y0 + row - 3, gx = col - 3;
    float v = 0.f;
    if (gy >= 0 && gy < 48 && gx >= 0 && gx < 48)
      v = frames[(bn * 4 + c) * HW + gy * 48 + gx];
    fs[c][row][col] = v;
  }
  __syncthreads();

  const int x   = tid % 48;
  const int yl  = (tid / 48) % 2;
  const int t   = tid / 96;                 // 0..2
  const int pix = (y0 + yl) * 48 + x;

  // weights for this t: kw[b,n,s,t,y,x]; s pairs positionally with K={7,5,3,1}
  float wgt[4];
#pragma unroll
  for (int s = 0; s < 4; ++s)
    wgt[s] = kw[bn * 27648 + s * 6912 + t * HW + pix];

  const int base0 = bn * (32 * QSTR) + t * HW + pix;   // + q*QSTR + c*3*HW
  float acc = 0.f;

  for (int c = 0; c < 4; ++c) {
    // pull the 7x7 window for this channel into registers
    float f[7][7];
#pragma unroll
    for (int i = 0; i < 7; ++i)
#pragma unroll
      for (int j = 0; j < 7; ++j)
        f[i][j] = fs[c][yl + i][x + j];

    const int bct = base0 + c * 3 * HW;
    float wsum;
    wsum  = wgt[0] * s_term<7, 0, 9>(core, bct, f);  // K=7, cur=9
    wsum += wgt[1] * s_term<5, 1, 4>(core, bct, f);  // K=5, cur=4
    wsum += wgt[2] * s_term<3, 2, 1>(core, bct, f);  // K=3, cur=1
    wsum += wgt[3] * s_term<1, 3, 0>(core, bct, f);  // K=1, cur=0
    acc += 0.25f * wsum;                             // mean over 4 branches
  }

  predI[(bn * 3 + t) * HW + pix] = acc;
}

// ---------------- Kernel 2: mean over N via WMMA ----------------
// pred[b,i] = sum_{n=0..7} (1/8) * P[b,n,i],  i in [0, 3*HW)
// A-operand (16x4 f32): lane L holds rows m=L&15, k-half kh=L>>4 (k={2kh,2kh+1}).
// Two chained V_WMMA_F32_16X16X4_F32 cover n=0..3 then n=4..7 (C accumulate).
// B is uniform 0.125 so every D column is identical: lanes 0-15 hold M=0..7 in
// VGPRs 0..7, lanes 16-31 hold M=8..15. Lanes 0 and 16 emit two b128 stores.
__global__ __launch_bounds__(128)
void mean_wmma_kernel(const float* __restrict__ P, float* __restrict__ outM) {
  const int lane  = threadIdx.x & 31;
  const int wave  = (blockIdx.x * 128 + threadIdx.x) >> 5;  // 0..1727
  const int b     = wave / 432;
  const int i0    = (wave % 432) * 16;

  const int m     = lane & 15;
  const int kh    = lane >> 4;
  const int pbase = b * (NB * 3 * HW) + i0 + m;

  v2f a0, a1;
  a0.x = P[pbase + (2 * kh + 0) * (3 * HW)];
  a0.y = P[pbase + (2 * kh + 1) * (3 * HW)];
  a1.x = P[pbase + (2 * kh + 4) * (3 * HW)];
  a1.y = P[pbase + (2 * kh + 5) * (3 * HW)];

  v2f bm;
  bm.x = 0.125f; bm.y = 0.125f;

  v8f c = {};
  c = __builtin_amdgcn_wmma_f32_16x16x4_f32(false, a0, false, bm, (short)0, c,
                                            false, false);
  c = __builtin_amdgcn_wmma_f32_16x16x4_f32(false, a1, false, bm, (short)0, c,
                                            false, false);

  // lane 0 -> out[i0+0..7], lane 16 -> out[i0+8..15]; 16-float aligned base
  if ((lane & 15) == 0) {
    const int obase = b * (3 * HW) + i0 + kh * 8;
    v4f lo, hi;
    lo.x = c[0]; lo.y = c[1]; lo.z = c[2]; lo.w = c[3];
    hi.x = c[4]; hi.y = c[5]; hi.z = c[6]; hi.w = c[7];
    *reinterpret_cast<v4f*>(outM + obase)     = lo;
    *reinterpret_cast<v4f*>(outM + obase + 4) = hi;
  }
}

// ---------------- launcher ----------------
extern "C" void kernel_launch(void* const* d_in, const int* in_sizes, int n_in,
                              void* d_out, int out_size, void* d_ws, size_t ws_size,
                              hipStream_t stream) {
  const float* frames = (const float*)d_in[0];
  const float* core   = (const float*)d_in[1];
  const float* kw     = (const float*)d_in[2];

  float* predI = (float*)d_out;                 // (B,N,3,48,48) = 221184
  float* predM = (float*)d_out + PREDI_ELEMS;   // (B,3,48,48)   =  27648

  kconv_kernel<<<NBATCH * NB * 24, 288, 0, stream>>>(frames, core, kw, predI);
  mean_wmma_kernel<<<432, 128, 0, stream>>>(predI, predM);
}